// Af_77515569758923
// MI455X (gfx1250) — compile-verified
//
#include <hip/hip_runtime.h>

// Bilinear affine warp (spatial transformer) for MI455X / gfx1250.
// N=128 frames, H=304, W=608, C=1, fp32.
//
// Strategy: one 16x16 pixel tile per wave32. The per-tile source-coordinate
// map x(py,px) = g(py) + f(px) is rank-1-separable, so it is produced by a
// single V_WMMA_F32_16X16X4_F32 per coordinate (2 WMMAs per tile), using the
// ISA-documented 32-bit A(16x4)/B(4x16)/D(16x16) lane layouts. The gather +
// blend inner loop is the HBM-bound part; WMMA (XDL, co-executes with VALU)
// makes coordinate generation effectively free. Output stores carry a
// non-temporal hint so the gather image stays resident in the 192MB L2.

typedef __attribute__((ext_vector_type(2))) float v2f;
typedef __attribute__((ext_vector_type(8))) float v8f;

#define IMG_W 608
#define IMG_H 304
#define N_FRAMES 128
#define TILES_X (IMG_W / 16)              // 38
#define TILES_Y (IMG_H / 16)              // 19
#define TILES_PER_FRAME (TILES_X * TILES_Y)   // 722
#define N_TILES (TILES_PER_FRAME * N_FRAMES)  // 92416
#define WAVES_PER_BLOCK 8
#define N_BLOCKS (N_TILES / WAVES_PER_BLOCK)  // 11552 (exact)

__global__ __launch_bounds__(256) void affine_bilinear_warp(
    const float* __restrict__ img,   // [N, H, W]
    const float* __restrict__ eye,   // [N, 2, 3]
    float* __restrict__ out)         // [N, H, W]
{
    const int wave = threadIdx.x >> 5;
    const int lane = threadIdx.x & 31;
    const int tile = blockIdx.x * WAVES_PER_BLOCK + wave;
    if (tile >= N_TILES) return;  // wave-uniform; EXEC stays all-ones

    const int frame = tile / TILES_PER_FRAME;
    const int trem  = tile - frame * TILES_PER_FRAME;
    const int ty    = trem / TILES_X;
    const int tx    = trem - ty * TILES_X;

    // 2x3 affine for this frame (uniform per wave; cache broadcasts).
    const float* e = eye + frame * 6;
    const float a00 = e[0], a01 = e[1], a02 = e[2];
    const float a10 = e[3], a11 = e[4], a12 = e[5];

    // Separable coordinate map, pre-scaled to pixel units:
    //   x_pix(py,px) = a00*xx(px)*W/2            +  (a01*yy(py)+a02+1)*W/2
    //                =      f_x(px)              +        g_x(py)
    //   y_pix(py,px) = a10*xx(px)*H/2            +  (a11*yy(py)+a12+1)*H/2
    // with xx(px) = px*2/(W-1)-1, yy(py) = py*2/(H-1)-1.
    const int   col = lane & 15;           // tile column this lane will own
    const int   hi  = lane >> 4;
    const float xxc = (float)(tx * 16 + col) * (2.0f / (IMG_W - 1)) - 1.0f;
    const float yym = (float)(ty * 16 + col) * (2.0f / (IMG_H - 1)) - 1.0f; // row index for A (lanes 0-15)
    const float fx  = a00 * xxc * (IMG_W * 0.5f);
    const float fy  = a10 * xxc * (IMG_H * 0.5f);
    const float gx  = (fmaf(a01, yym, a02) + 1.0f) * (IMG_W * 0.5f);
    const float gy  = (fmaf(a11, yym, a12) + 1.0f) * (IMG_H * 0.5f);

    // A (16x4 f32, 2 VGPRs): lanes 0-15 -> {K0,K1}, lanes 16-31 -> {K2,K3}=0.
    //   A[m][0] = g(m), A[m][1] = 1.
    // B (4x16 f32, 2 VGPRs): v0 lanes0-15 = row K0, v1 lanes0-15 = row K1,
    //   hi-lanes = rows K2,K3 = 0.   B[0][n] = 1, B[1][n] = f(n).
    const bool lo = (hi == 0);
    v2f ax, bx, ay, by;
    ax.x = lo ? gx : 0.0f;   ax.y = lo ? 1.0f : 0.0f;
    bx.x = lo ? 1.0f : 0.0f; bx.y = lo ? fx   : 0.0f;
    ay.x = lo ? gy : 0.0f;   ay.y = lo ? 1.0f : 0.0f;
    by.x = lo ? 1.0f : 0.0f; by.y = lo ? fy   : 0.0f;

    v8f cz = {};
    // D[m][n] = g(m) + f(n): the 16x16 source-coordinate map for this tile.
    v8f xmap = __builtin_amdgcn_wmma_f32_16x16x4_f32(
        false, ax, false, bx, (short)0, cz, false, false);
    v8f ymap = __builtin_amdgcn_wmma_f32_16x16x4_f32(
        false, ay, false, by, (short)0, cz, false, false);

    // C/D layout: lane (0-15): N=lane, M=v;  lane (16-31): N=lane-16, M=v+8.
    const int px = tx * 16 + col;
    const float* ib = img + (size_t)frame * (IMG_H * IMG_W);
    float* ob = out + (size_t)frame * (IMG_H * IMG_W) + px;

#pragma unroll
    for (int r = 0; r < 8; ++r) {
        const int py = ty * 16 + r + 8 * hi;
        const float x = xmap[r];
        const float y = ymap[r];

        const float xf = floorf(x);
        const float yf = floorf(y);
        int x0 = (int)xf;
        int y0 = (int)yf;
        int x1 = x0 + 1;
        int y1 = y0 + 1;
        x0 = min(max(x0, 0), IMG_W - 1);
        x1 = min(max(x1, 0), IMG_W - 1);
        y0 = min(max(y0, 0), IMG_H - 1);
        y1 = min(max(y1, 0), IMG_H - 1);

        const float Ia = ib[y0 * IMG_W + x0];
        const float Ib = ib[y1 * IMG_W + x0];
        const float Ic = ib[y0 * IMG_W + x1];
        const float Id = ib[y1 * IMG_W + x1];

        const float x0f = (float)x0, x1f = (float)x1;
        const float y0f = (float)y0, y1f = (float)y1;
        const float wa = (x1f - x) * (y1f - y);
        const float wb = (x1f - x) * (y - y0f);
        const float wc = (x - x0f) * (y1f - y);
        const float wd = (x - x0f) * (y - y0f);

        float v = wa * Ia;
        v = fmaf(wb, Ib, v);
        v = fmaf(wc, Ic, v);
        v = fmaf(wd, Id, v);

        // NT store: output is written once; keep the gather image in L2.
        __builtin_nontemporal_store(v, ob + py * IMG_W);
    }
}

extern "C" void kernel_launch(void* const* d_in, const int* in_sizes, int n_in,
                              void* d_out, int out_size, void* d_ws, size_t ws_size,
                              hipStream_t stream) {
    (void)in_sizes; (void)n_in; (void)out_size; (void)d_ws; (void)ws_size;
    const float* stimuli = (const float*)d_in[0];  // [4,32,304,608,1] fp32
    const float* eye     = (const float*)d_in[1];  // [4,32,2,3]       fp32
    float* out           = (float*)d_out;          // [4,32,304,608,1] fp32

    affine_bilinear_warp<<<N_BLOCKS, 256, 0, stream>>>(stimuli, eye, out);
}